// MessagePassing_35536559407204
// MI455X (gfx1250) — compile-verified
//
#include <hip/hip_runtime.h>

// ---------------------------------------------------------------------------
// segment_sum:  out[col[e]] += x[row[e]],  E=625000, d=128, N=100000.
//
// MI455X reasoning: working set (x 51.2MB + out 51.2MB + edges 10MB) is
// L2-resident (192MB). Naive scatter = 80M f32 atomics -> bound by L2 atomic
// op rate, ~20x above the L2-bandwidth floor. Instead: counting-sort edges by
// destination (1.25M cheap u32 atomics + LDS scans), then one wave32 per node
// accumulates its segment in registers and does a single plain 512B store.
// Gather pass uses cooperative index fetch: one coalesced load brings up to
// 32 segment indices into the wave; v_readlane broadcasts them (no per-edge
// index-load latency chain). Fallback to atomic scatter if d_ws too small.
// ---------------------------------------------------------------------------

typedef unsigned int u32;

constexpr int FEAT            = 128;   // floats per node row
constexpr int WAVES_PER_BLOCK = 8;     // 256 threads = 8 wave32
constexpr int BLOCK           = WAVES_PER_BLOCK * 32;
constexpr int SCAN_CHUNK      = 1024;  // elements per scan block (256 thr x 4)

// ======================= sorted (binned) path ==============================

__global__ __launch_bounds__(256) void zero_u32_kernel(u32* __restrict__ p, int n) {
  const int i = blockIdx.x * blockDim.x + threadIdx.x;
  if (i < n) p[i] = 0u;
}

// counts[col[e]]++  (u32 atomics, random over 400KB -> L2 resident)
__global__ __launch_bounds__(256) void count_kernel(const int* __restrict__ col,
                                                    u32* __restrict__ counts, int E) {
  const int e = blockIdx.x * blockDim.x + threadIdx.x;
  if (e >= E) return;
  const int c = __builtin_nontemporal_load(col + e);
  atomicAdd(&counts[c], 1u);
}

// per-block sum of a 1024-element chunk of counts
__global__ __launch_bounds__(256) void scan_partial_kernel(const u32* __restrict__ counts,
                                                           u32* __restrict__ block_sums,
                                                           int N) {
  __shared__ u32 s[256];
  const int t    = threadIdx.x;
  const int base = blockIdx.x * SCAN_CHUNK + t * 4;
  u32 sum = 0;
#pragma unroll
  for (int j = 0; j < 4; ++j) {
    const int idx = base + j;
    if (idx < N) sum += counts[idx];
  }
  s[t] = sum;
  __syncthreads();
#pragma unroll
  for (int off = 128; off > 0; off >>= 1) {
    if (t < off) s[t] += s[t + off];
    __syncthreads();
  }
  if (t == 0) block_sums[blockIdx.x] = s[0];
}

// exclusive scan of block sums (single block, up to 1024 blocks -> N <= 1M)
__global__ __launch_bounds__(1024) void scan_blocksums_kernel(u32* __restrict__ block_sums,
                                                              int nb) {
  __shared__ u32 s[1024];
  const int t = threadIdx.x;
  const u32 orig = (t < nb) ? block_sums[t] : 0u;
  s[t] = orig;
  __syncthreads();
#pragma unroll
  for (int off = 1; off < 1024; off <<= 1) {
    const u32 add = (t >= off) ? s[t - off] : 0u;
    __syncthreads();
    s[t] += add;
    __syncthreads();
  }
  if (t < nb) block_sums[t] = s[t] - orig;  // exclusive
}

// exclusive scan within each chunk + block base -> offsets[] and cursor[]
__global__ __launch_bounds__(256) void scan_final_kernel(const u32* __restrict__ counts,
                                                         const u32* __restrict__ block_sums,
                                                         u32* __restrict__ offsets,
                                                         u32* __restrict__ cursor,
                                                         int N, int E) {
  __shared__ u32 s[256];
  const int t    = threadIdx.x;
  const int base = blockIdx.x * SCAN_CHUNK + t * 4;
  u32 v[4];
  u32 tsum = 0;
#pragma unroll
  for (int j = 0; j < 4; ++j) {
    const int idx = base + j;
    v[j] = (idx < N) ? counts[idx] : 0u;
    tsum += v[j];
  }
  s[t] = tsum;
  __syncthreads();
#pragma unroll
  for (int off = 1; off < 256; off <<= 1) {
    const u32 add = (t >= off) ? s[t - off] : 0u;
    __syncthreads();
    s[t] += add;
    __syncthreads();
  }
  u32 run = block_sums[blockIdx.x] + (s[t] - tsum);  // exclusive within grid
#pragma unroll
  for (int j = 0; j < 4; ++j) {
    const int idx = base + j;
    if (idx < N) {
      offsets[idx] = run;
      cursor[idx]  = run;
      run += v[j];
    }
  }
  if (blockIdx.x == 0 && t == 0) offsets[N] = (u32)E;
}

// rows_sorted[cursor[col[e]]++] = row[e]
__global__ __launch_bounds__(256) void reorder_kernel(const int* __restrict__ row,
                                                      const int* __restrict__ col,
                                                      u32* __restrict__ cursor,
                                                      u32* __restrict__ rows_sorted, int E) {
  const int e = blockIdx.x * blockDim.x + threadIdx.x;
  if (e >= E) return;
  const int c = __builtin_nontemporal_load(col + e);
  const int r = __builtin_nontemporal_load(row + e);
  const u32 pos = atomicAdd(&cursor[c], 1u);
  rows_sorted[pos] = (u32)r;
}

// one wave32 per node: cooperative index fetch (1 coalesced load for up to 32
// segment indices), v_readlane broadcast, register accumulation, single
// coalesced 512B store. No f32 atomics, no pre-zeroing of out.
__global__ __launch_bounds__(BLOCK) void gather_segment_kernel(
    const float* __restrict__ x,
    const u32*   __restrict__ offsets,
    const u32*   __restrict__ rows_sorted,
    float*       __restrict__ out,
    int N)
{
  const int lane = threadIdx.x & 31;
  const int node = blockIdx.x * WAVES_PER_BLOCK + (threadIdx.x >> 5);
  if (node >= N) return;  // wave-uniform: node is identical across the wave

  // lanes 0/1 fetch begin/end in a single vmem op, broadcast via readlane
  const u32 off2 = offsets[node + (lane & 1)];
  const u32 beg  = (u32)__builtin_amdgcn_readlane((int)off2, 0);
  const u32 end  = (u32)__builtin_amdgcn_readlane((int)off2, 1);
  const u32 cnt  = end - beg;

  // cooperative fetch: lane l holds segment index l (covers deg <= 32, which
  // is essentially always true for Poisson(6.25) degrees)
  int idx_l = 0;
  if (lane < cnt) idx_l = (int)rows_sorted[beg + lane];

  float4 acc = make_float4(0.f, 0.f, 0.f, 0.f);
  const u32 c32 = cnt < 32u ? cnt : 32u;
  u32 k = 0;
  // unroll by 2: two independent 512B gathers in flight, indices from VALU
  for (; k + 1 < c32; k += 2) {
    const int r0 = __builtin_amdgcn_readlane(idx_l, (int)k);
    const int r1 = __builtin_amdgcn_readlane(idx_l, (int)(k + 1));
    const float4 a = ((const float4*)(x + (size_t)r0 * FEAT))[lane];
    const float4 b = ((const float4*)(x + (size_t)r1 * FEAT))[lane];
    acc.x += a.x; acc.y += a.y; acc.z += a.z; acc.w += a.w;
    acc.x += b.x; acc.y += b.y; acc.z += b.z; acc.w += b.w;
  }
  if (k < c32) {
    const int r0 = __builtin_amdgcn_readlane(idx_l, (int)k);
    const float4 a = ((const float4*)(x + (size_t)r0 * FEAT))[lane];
    acc.x += a.x; acc.y += a.y; acc.z += a.z; acc.w += a.w;
  }
  // rare tail for degree > 32
  for (u32 i = beg + 32u; i < end; ++i) {
    const int r = __builtin_amdgcn_readfirstlane((int)rows_sorted[i]);
    const float4 a = ((const float4*)(x + (size_t)r * FEAT))[lane];
    acc.x += a.x; acc.y += a.y; acc.z += a.z; acc.w += a.w;
  }

  ((float4*)(out + (size_t)node * FEAT))[lane] = acc;
}

// ======================= atomic fallback path ==============================

__global__ __launch_bounds__(256) void zero_out_kernel(float* __restrict__ out, int n) {
  const int n4 = n >> 2;
  const int i  = blockIdx.x * blockDim.x + threadIdx.x;
  if (i < n4) ((float4*)out)[i] = make_float4(0.f, 0.f, 0.f, 0.f);
  const int tail = n - (n4 << 2);
  if (i < tail) out[(n4 << 2) + i] = 0.f;
}

constexpr int EPW = 4;  // edges per wave (ILP) in fallback

__global__ __launch_bounds__(BLOCK) void scatter_add_kernel(
    const float* __restrict__ x,
    const int*   __restrict__ row_idx,
    const int*   __restrict__ col_idx,
    float*       __restrict__ out,
    int E)
{
  const int lane = threadIdx.x & 31;
  const int wave = blockIdx.x * WAVES_PER_BLOCK + (threadIdx.x >> 5);
  const int e0   = wave * EPW;
  if (e0 >= E) return;

  if (lane == 0) {
    const int pf = e0 + 2 * WAVES_PER_BLOCK * EPW;
    if (pf < E) {
      __builtin_prefetch(row_idx + pf, 0, 0);
      __builtin_prefetch(col_idx + pf, 0, 0);
    }
  }

  int rows[EPW], cols[EPW];
#pragma unroll
  for (int k = 0; k < EPW; ++k) {
    const int e = e0 + k;
    if (e < E) {
      rows[k] = __builtin_amdgcn_readfirstlane(__builtin_nontemporal_load(row_idx + e));
      cols[k] = __builtin_amdgcn_readfirstlane(__builtin_nontemporal_load(col_idx + e));
    } else {
      rows[k] = 0;
      cols[k] = -1;
    }
  }

  float4 v[EPW];
#pragma unroll
  for (int k = 0; k < EPW; ++k)
    if (cols[k] >= 0) v[k] = ((const float4*)(x + (size_t)rows[k] * FEAT))[lane];

#pragma unroll
  for (int k = 0; k < EPW; ++k) {
    if (cols[k] >= 0) {
      float* dst = out + (size_t)cols[k] * FEAT + lane * 4;
      unsafeAtomicAdd(dst + 0, v[k].x);
      unsafeAtomicAdd(dst + 1, v[k].y);
      unsafeAtomicAdd(dst + 2, v[k].z);
      unsafeAtomicAdd(dst + 3, v[k].w);
    }
  }
}

// =========================== launcher ======================================

extern "C" void kernel_launch(void* const* d_in, const int* in_sizes, int n_in,
                              void* d_out, int out_size, void* d_ws, size_t ws_size,
                              hipStream_t stream) {
  const float* x  = (const float*)d_in[0];
  const int*   ei = (const int*)d_in[1];       // edge_index flat [2, E]
  const int E = in_sizes[1] / 2;
  const int N = in_sizes[0] / FEAT;
  const int* row_idx = ei;
  const int* col_idx = ei + E;
  float* out = (float*)d_out;

  const int nb = (N + SCAN_CHUNK - 1) / SCAN_CHUNK;  // scan blocks

  // workspace layout (u32): offsets[N+1] | cursor[N] | rows_sorted[E] |
  //                         block_sums[1024] | counts[N]
  const size_t need_u32 = (size_t)(N + 1) + N + E + 1024 + N;
  const bool use_sorted = (ws_size >= need_u32 * sizeof(u32)) && (nb <= 1024);

  if (use_sorted) {
    u32* ws          = (u32*)d_ws;
    u32* offsets     = ws;
    u32* cursor      = offsets + (N + 1);
    u32* rows_sorted = cursor + N;
    u32* block_sums  = rows_sorted + E;
    u32* counts      = block_sums + 1024;

    const int T = 256;
    zero_u32_kernel<<<(N + T - 1) / T, T, 0, stream>>>(counts, N);
    count_kernel<<<(E + T - 1) / T, T, 0, stream>>>(col_idx, counts, E);
    scan_partial_kernel<<<nb, T, 0, stream>>>(counts, block_sums, N);
    scan_blocksums_kernel<<<1, 1024, 0, stream>>>(block_sums, nb);
    scan_final_kernel<<<nb, T, 0, stream>>>(counts, block_sums, offsets, cursor, N, E);
    reorder_kernel<<<(E + T - 1) / T, T, 0, stream>>>(row_idx, col_idx, cursor,
                                                      rows_sorted, E);
    const int gblocks = (N + WAVES_PER_BLOCK - 1) / WAVES_PER_BLOCK;
    gather_segment_kernel<<<gblocks, BLOCK, 0, stream>>>(x, offsets, rows_sorted, out, N);
  } else {
    // fallback: direct atomic scatter-add
    const int n4 = out_size >> 2;
    zero_out_kernel<<<(n4 + 255) / 256, 256, 0, stream>>>(out, out_size);
    const int waves  = (E + EPW - 1) / EPW;
    const int blocks = (waves + WAVES_PER_BLOCK - 1) / WAVES_PER_BLOCK;
    scatter_add_kernel<<<blocks, BLOCK, 0, stream>>>(x, row_idx, col_idx, out, E);
  }
}